// GAT_56702158241981
// MI455X (gfx1250) — compile-verified
//
#include <hip/hip_runtime.h>
#include <hip/hip_bf16.h>

// ---------------------------------------------------------------------------
// GAT (2x GATConv + mean-pool + linear) for MI455X / gfx1250, wave32.
// Dense GEMMs use V_WMMA_F32_16X16X4_F32 (fp32 in/out, matches reference
// precision). Edge softmax/scatter phases are L2-resident atomic passes.
// Round 2: branch-free A-fragment loads (clamp + cndmask, global_load_b64)
// and pre-swizzled B staging in LDS (one ds_load_b64 per fragment).
// ---------------------------------------------------------------------------

typedef float v2f __attribute__((ext_vector_type(2)));
typedef float v8f __attribute__((ext_vector_type(8)));

#define IN_C   128
#define HID_ALL 128   // HEADS(4) * HID_C(32)
#define OUT_C  64
#define NGRAPH 64
#define NCLS   2
#define NEG_SLOPE 0.2f

// ---- helpers --------------------------------------------------------------

__device__ __forceinline__ unsigned fenc_gat(float f) {
    unsigned u = __float_as_uint(f);
    return (u & 0x80000000u) ? ~u : (u | 0x80000000u);   // order-preserving map
}
__device__ __forceinline__ float fdec_gat(unsigned u) {
    return (u & 0x80000000u) ? __uint_as_float(u ^ 0x80000000u)
                             : __uint_as_float(~u);
}
__device__ __forceinline__ float lrelu_gat(float v) {
    return v > 0.f ? v : NEG_SLOPE * v;
}
__device__ __forceinline__ float elu_gat(float v) {
    return v > 0.f ? v : (__expf(v) - 1.f);
}
// edge e in [0, E2): first E are real edges (edge_index rows), rest self-loops
__device__ __forceinline__ void edge_sd(const int* __restrict__ ei, int E,
                                        int e, int& s, int& d) {
    if (e < E) { s = ei[e]; d = ei[E + e]; }
    else       { s = d = e - E; }
}

// ---- workspace zeroing ----------------------------------------------------

__global__ void gat_zero_kernel(float* __restrict__ p, size_t n) {
    size_t i = (size_t)blockIdx.x * blockDim.x + threadIdx.x;
    if (i < n) p[i] = 0.f;
}

// ---- WMMA GEMM: C[nrows x NCOLS] = A[nrows x K] * B[K x NCOLS] ------------
// Block = 256 threads = 8 wave32s; wave w computes rows [blk*128+16w, +16).
// B staged in LDS pre-swizzled into per-lane v2f fragments:
//   element (k, n), k = 4*kg + 2*hl + j, lane = hl*16 + (n&15), t = n>>4
//   -> Bs[kg*4*NCOLS + t*64 + lane*2 + j]
// so the inner loop is a single ds_load_b64 per (kg, t).

template <int K, int NCOLS>
__global__ __launch_bounds__(256) void gat_gemm_wmma(
    const float* __restrict__ A, const float* __restrict__ Bg,
    float* __restrict__ C, int nrows)
{
    constexpr int NT  = NCOLS / 16;
    constexpr int NKG = K / 4;
    __shared__ float Bs[K * NCOLS];
    for (int i = threadIdx.x; i < K * NCOLS; i += 256) {
        int k = i / NCOLS, n = i - k * NCOLS;
        int kg = k >> 2, rem = k & 3;
        int ln = ((rem >> 1) << 4) | (n & 15);
        Bs[kg * (4 * NCOLS) + (n >> 4) * 64 + ln * 2 + (rem & 1)] = Bg[i];
    }
    __syncthreads();

    const int wave = threadIdx.x >> 5;
    const int lane = threadIdx.x & 31;
    const int hl   = lane >> 4;        // half-wave: 0 or 1
    const int l16  = lane & 15;
    const int rowbase = blockIdx.x * 128 + wave * 16;
    const int arow = rowbase + l16;
    const bool aval = (arow < nrows);
    const int crow = aval ? arow : (nrows - 1);          // clamped: always legal
    const float* __restrict__ Arow = A + (size_t)crow * K + 2 * hl;
    const float* __restrict__ Bl   = Bs + lane * 2;

    v8f acc[NT] = {};

    for (int kg = 0; kg < NKG; ++kg) {
        v2f a = *(const v2f*)(Arow + kg * 4);            // global_load_b64
        a.x = aval ? a.x : 0.f;                          // v_cndmask, no branch
        a.y = aval ? a.y : 0.f;
#pragma unroll
        for (int t = 0; t < NT; ++t) {
            v2f b = *(const v2f*)(Bl + kg * (4 * NCOLS) + t * 64);  // ds_load_b64
            acc[t] = __builtin_amdgcn_wmma_f32_16x16x4_f32(
                false, a, false, b, (short)0, acc[t], false, false);
        }
    }

#pragma unroll
    for (int t = 0; t < NT; ++t) {
#pragma unroll
        for (int r = 0; r < 8; ++r) {         // C/D: (M = r + 8*hl, N = l16)
            int row = rowbase + hl * 8 + r;
            if (row < nrows)
                C[(size_t)row * NCOLS + t * 16 + l16] = acc[t][r];
        }
    }
}

// ---- per-node attention dots ---------------------------------------------
// conv1: 128 threads per node = 4 wave32s = 4 heads; full wave reduction.

__global__ void gat_node_att1(const float* __restrict__ h,
                              const float* __restrict__ atts,
                              const float* __restrict__ attd,
                              float* __restrict__ as, float* __restrict__ ad,
                              int n)
{
    int node = blockIdx.x * 2 + (threadIdx.x >> 7);
    if (node >= n) return;
    int c = threadIdx.x & 127;
    float hv = h[(size_t)node * HID_ALL + c];
    float vs = hv * atts[c];
    float vd = hv * attd[c];
#pragma unroll
    for (int o = 16; o > 0; o >>= 1) {
        vs += __shfl_xor(vs, o, 32);
        vd += __shfl_xor(vd, o, 32);
    }
    if ((c & 31) == 0) {
        int head = c >> 5;
        as[node * 4 + head] = vs;
        ad[node * 4 + head] = vd;
    }
}

// conv2: one wave per node, 64 channels in two chunks.
__global__ void gat_node_att2(const float* __restrict__ h,
                              const float* __restrict__ atts,
                              const float* __restrict__ attd,
                              float* __restrict__ as, float* __restrict__ ad,
                              int n)
{
    int node = blockIdx.x * 8 + (threadIdx.x >> 5);
    if (node >= n) return;
    int lane = threadIdx.x & 31;
    const float* hp = h + (size_t)node * OUT_C;
    float vs = hp[lane] * atts[lane] + hp[lane + 32] * atts[lane + 32];
    float vd = hp[lane] * attd[lane] + hp[lane + 32] * attd[lane + 32];
#pragma unroll
    for (int o = 16; o > 0; o >>= 1) {
        vs += __shfl_xor(vs, o, 32);
        vd += __shfl_xor(vd, o, 32);
    }
    if (lane == 0) { as[node] = vs; ad[node] = vd; }
}

// ---- edge passes: conv1 (4 heads) ----------------------------------------

__global__ void gat_edge_max1(const int* __restrict__ ei, int E, int E2,
                              const float* __restrict__ as,
                              const float* __restrict__ ad,
                              unsigned* __restrict__ m)
{
    int e = blockIdx.x * blockDim.x + threadIdx.x;
    if (e >= E2) return;
    int s, d; edge_sd(ei, E, e, s, d);
#pragma unroll
    for (int h = 0; h < 4; ++h) {
        float l = lrelu_gat(as[s * 4 + h] + ad[d * 4 + h]);
        atomicMax(&m[d * 4 + h], fenc_gat(l));
    }
}

__global__ void gat_edge_expsum1(const int* __restrict__ ei, int E, int E2,
                                 const float* __restrict__ as,
                                 const float* __restrict__ ad,
                                 const unsigned* __restrict__ m,
                                 float* __restrict__ alpha,
                                 float* __restrict__ ssum)
{
    int e = blockIdx.x * blockDim.x + threadIdx.x;
    if (e >= E2) return;
    int s, d; edge_sd(ei, E, e, s, d);
#pragma unroll
    for (int h = 0; h < 4; ++h) {
        float l  = lrelu_gat(as[s * 4 + h] + ad[d * 4 + h]);
        float mm = fdec_gat(m[d * 4 + h]);
        float ex = __expf(l - mm);
        alpha[(size_t)e * 4 + h] = ex;
        atomicAdd(&ssum[d * 4 + h], ex);
    }
}

// wave per edge; lane = channel within head; 4 heads looped -> coalesced
// 32-lane global_atomic_add_f32 bursts, all landing in L2 (agg fits in 192MB).
__global__ void gat_edge_scatter1(const int* __restrict__ ei, int E, int E2,
                                  const float* __restrict__ h,
                                  const float* __restrict__ alpha,
                                  const float* __restrict__ ssum,
                                  float* __restrict__ agg)
{
    int e    = (blockIdx.x * blockDim.x + threadIdx.x) >> 5;
    int lane = threadIdx.x & 31;
    if (e >= E2) return;
    int s, d; edge_sd(ei, E, e, s, d);
    const float* hs = h + (size_t)s * HID_ALL;
    float* ag = agg + (size_t)d * HID_ALL;
#pragma unroll
    for (int hh = 0; hh < 4; ++hh) {
        float a = alpha[(size_t)e * 4 + hh] / (ssum[d * 4 + hh] + 1e-16f);
        atomicAdd(&ag[hh * 32 + lane], hs[hh * 32 + lane] * a);
    }
}

// ---- edge passes: conv2 (1 head, 64 ch) ----------------------------------

__global__ void gat_edge_max2(const int* __restrict__ ei, int E, int E2,
                              const float* __restrict__ as,
                              const float* __restrict__ ad,
                              unsigned* __restrict__ m)
{
    int e = blockIdx.x * blockDim.x + threadIdx.x;
    if (e >= E2) return;
    int s, d; edge_sd(ei, E, e, s, d);
    atomicMax(&m[d], fenc_gat(lrelu_gat(as[s] + ad[d])));
}

__global__ void gat_edge_expsum2(const int* __restrict__ ei, int E, int E2,
                                 const float* __restrict__ as,
                                 const float* __restrict__ ad,
                                 const unsigned* __restrict__ m,
                                 float* __restrict__ alpha,
                                 float* __restrict__ ssum)
{
    int e = blockIdx.x * blockDim.x + threadIdx.x;
    if (e >= E2) return;
    int s, d; edge_sd(ei, E, e, s, d);
    float l  = lrelu_gat(as[s] + ad[d]);
    float ex = __expf(l - fdec_gat(m[d]));
    alpha[e] = ex;
    atomicAdd(&ssum[d], ex);
}

__global__ void gat_edge_scatter2(const int* __restrict__ ei, int E, int E2,
                                  const float* __restrict__ h,
                                  const float* __restrict__ alpha,
                                  const float* __restrict__ ssum,
                                  float* __restrict__ agg)
{
    int e    = (blockIdx.x * blockDim.x + threadIdx.x) >> 5;
    int lane = threadIdx.x & 31;
    if (e >= E2) return;
    int s, d; edge_sd(ei, E, e, s, d);
    float a = alpha[e] / (ssum[d] + 1e-16f);
    const float* hs = h + (size_t)s * OUT_C;
    float* ag = agg + (size_t)d * OUT_C;
    atomicAdd(&ag[lane],      hs[lane]      * a);
    atomicAdd(&ag[lane + 32], hs[lane + 32] * a);
}

// ---- epilogues ------------------------------------------------------------

__global__ void gat_finalize1(const float* __restrict__ agg,
                              const float* __restrict__ b,
                              float* __restrict__ out, size_t total)
{
    size_t i = (size_t)blockIdx.x * blockDim.x + threadIdx.x;
    if (i >= total) return;
    out[i] = elu_gat(agg[i] + b[i & (HID_ALL - 1)]);
}

__global__ void gat_finalize2_pool(const float* __restrict__ agg,
                                   const float* __restrict__ b,
                                   const int* __restrict__ batch,
                                   float* __restrict__ pooled,
                                   float* __restrict__ cnt, int n)
{
    int i = blockIdx.x * blockDim.x + threadIdx.x;
    if (i >= n * OUT_C) return;
    int node = i >> 6, c = i & 63;
    float v = elu_gat(agg[i] + b[c]);
    int g = batch[node];
    atomicAdd(&pooled[g * OUT_C + c], v);
    if (c == 0) atomicAdd(&cnt[g], 1.f);
}

__global__ void gat_classify(const float* __restrict__ pooled,
                             const float* __restrict__ cnt,
                             const float* __restrict__ Wc,
                             const float* __restrict__ bc,
                             float* __restrict__ out)
{
    int t = threadIdx.x;                // 128 = NGRAPH*NCLS
    int g = t >> 1, k = t & 1;
    float inv = 1.f / fmaxf(cnt[g], 1.f);
    float acc = bc[k];
    for (int c = 0; c < OUT_C; ++c)
        acc += pooled[g * OUT_C + c] * inv * Wc[c * NCLS + k];
    out[g * NCLS + k] = acc;
}

// ---- launch ---------------------------------------------------------------

extern "C" void kernel_launch(void* const* d_in, const int* in_sizes, int n_in,
                              void* d_out, int out_size, void* d_ws,
                              size_t ws_size, hipStream_t stream)
{
    const float* x     = (const float*)d_in[0];
    const int*   ei    = (const int*)  d_in[1];
    const int*   batch = (const int*)  d_in[2];
    const float* W1    = (const float*)d_in[3];
    const float* atts1 = (const float*)d_in[4];
    const float* attd1 = (const float*)d_in[5];
    const float* b1    = (const float*)d_in[6];
    const float* W2    = (const float*)d_in[7];
    const float* atts2 = (const float*)d_in[8];
    const float* attd2 = (const float*)d_in[9];
    const float* b2    = (const float*)d_in[10];
    const float* Wc    = (const float*)d_in[11];
    const float* bc    = (const float*)d_in[12];
    float* out = (float*)d_out;

    const int N  = in_sizes[0] / IN_C;
    const int E  = in_sizes[1] / 2;
    const int E2 = E + N;               // + self loops

    // ---- workspace layout (floats) ----
    float* base = (float*)d_ws;
    size_t off = 0;
    auto take = [&](size_t c) { float* p = base + off; off += c; return p; };

    float* h1     = take((size_t)N * HID_ALL);   // conv1 pre-agg features, later h1'
    float* as1    = take((size_t)N * 4);
    float* ad1    = take((size_t)N * 4);
    float* as2    = take((size_t)N);
    float* ad2    = take((size_t)N);
    float* alpha1 = take((size_t)E2 * 4);        // alpha2 aliases this region

    float* zbeg   = base + off;                  // ---- zeroed every call ----
    float* agg1   = take((size_t)N * HID_ALL);   // reused as h2 after finalize1
    float* agg2   = take((size_t)N * OUT_C);
    float* m1f    = take((size_t)N * 4);
    float* s1     = take((size_t)N * 4);
    float* m2f    = take((size_t)N);
    float* s2     = take((size_t)N);
    float* pooled = take((size_t)NGRAPH * OUT_C);
    float* cnt    = take((size_t)NGRAPH);
    size_t zcount = (size_t)((base + off) - zbeg);

    float* h2     = agg1;
    float* alpha2 = alpha1;

    const int TB = 256;
    // 0) zero atomic accumulators (poisoned to 0xAA by harness, not restored)
    gat_zero_kernel<<<(unsigned)((zcount + TB - 1) / TB), TB, 0, stream>>>(zbeg, zcount);

    // 1) conv1 GEMM: h1 = x @ W1  [N,128]x[128,128]
    gat_gemm_wmma<IN_C, HID_ALL><<<(unsigned)((N + 127) / 128), TB, 0, stream>>>(x, W1, h1, N);
    // 2) per-node attention terms
    gat_node_att1<<<(unsigned)((N + 1) / 2), TB, 0, stream>>>(h1, atts1, attd1, as1, ad1, N);
    // 3) segment softmax + scatter
    unsigned egrid = (unsigned)((E2 + TB - 1) / TB);
    unsigned sgrid = (unsigned)(((size_t)E2 * 32 + TB - 1) / TB);
    gat_edge_max1   <<<egrid, TB, 0, stream>>>(ei, E, E2, as1, ad1, (unsigned*)m1f);
    gat_edge_expsum1<<<egrid, TB, 0, stream>>>(ei, E, E2, as1, ad1, (const unsigned*)m1f, alpha1, s1);
    gat_edge_scatter1<<<sgrid, TB, 0, stream>>>(ei, E, E2, h1, alpha1, s1, agg1);
    // 4) h1' = elu(agg1 + b1)  (overwrite h1)
    size_t t1 = (size_t)N * HID_ALL;
    gat_finalize1<<<(unsigned)((t1 + TB - 1) / TB), TB, 0, stream>>>(agg1, b1, h1, t1);

    // 5) conv2 GEMM: h2 = h1' @ W2  [N,128]x[128,64]  (h2 aliases agg1)
    gat_gemm_wmma<HID_ALL, OUT_C><<<(unsigned)((N + 127) / 128), TB, 0, stream>>>(h1, W2, h2, N);
    gat_node_att2<<<(unsigned)((N + 7) / 8), TB, 0, stream>>>(h2, atts2, attd2, as2, ad2, N);
    gat_edge_max2   <<<egrid, TB, 0, stream>>>(ei, E, E2, as2, ad2, (unsigned*)m2f);
    gat_edge_expsum2<<<egrid, TB, 0, stream>>>(ei, E, E2, as2, ad2, (const unsigned*)m2f, alpha2, s2);
    gat_edge_scatter2<<<sgrid, TB, 0, stream>>>(ei, E, E2, h2, alpha2, s2, agg2);

    // 6) elu + global mean pool, then classifier
    gat_finalize2_pool<<<(unsigned)(((size_t)N * OUT_C + TB - 1) / TB), TB, 0, stream>>>(
        agg2, b2, batch, pooled, cnt, N);
    gat_classify<<<1, NGRAPH * NCLS, 0, stream>>>(pooled, cnt, Wc, bc, out);
}